// NGCF_4449586119373
// MI455X (gfx1250) — compile-verified
//
#include <hip/hip_runtime.h>
#include <hip/hip_bf16.h>
#include <stdint.h>

#define NUM_USER   1000000
#define NUM_ITEM   500000
#define EMBED_DIM  64
#define NUM_EDGES  2000000

#define TILE_EDGES    1024
#define SCATTER_BLOCK 256

// native clang vector type (NOT HIP's int4 struct) — matches the builtin's
// expected parameter type '__attribute__((__vector_size__(16))) int *'
typedef int v4i __attribute__((vector_size(16)));

// ---------------------------------------------------------------------------
// CDNA5 async global->LDS helpers (gfx1250). Guarded so host pass and older
// toolchains still compile (fallback = plain vector copy through VGPRs).
// ---------------------------------------------------------------------------
__device__ __forceinline__ void async_copy_b128(const int* gsrc, int* ldst) {
#if defined(__gfx1250__) && __has_builtin(__builtin_amdgcn_global_load_async_to_lds_b128)
    __builtin_amdgcn_global_load_async_to_lds_b128(
        (v4i*)gsrc,             // global source (cast away const)
        (v4i*)ldst,             // LDS destination (generic ptr into __shared__)
        /*offset=*/0, /*cpol=*/0);
#else
    *(v4i*)ldst = *(const v4i*)gsrc;
#endif
}

__device__ __forceinline__ void async_wait_all() {
#if defined(__gfx1250__) && __has_builtin(__builtin_amdgcn_global_load_async_to_lds_b128)
#if __has_builtin(__builtin_amdgcn_s_wait_asynccnt)
    __builtin_amdgcn_s_wait_asynccnt(0);
#else
    asm volatile("s_wait_asynccnt 0" ::: "memory");
#endif
#endif
}

// ---------------------------------------------------------------------------
// Per-edge body: 16 lanes cooperate on one edge; lane `sub` owns one float4 of
// the 64-float row -> coalesced 256B gather, then 4 no-return
// global_atomic_add_f32 into the L2-resident accumulator.
// ---------------------------------------------------------------------------
__device__ __forceinline__ void process_edge(
    const float* __restrict__ user, float* __restrict__ out,
    unsigned* __restrict__ counts, const int* s_src, const int* s_dst,
    int le, int sub, bool do_prefetch) {
    const int src = s_src[le];
    const int dst = s_dst[le];

    // prefetch a future gather row (global_prefetch_b8); one line per 8 lanes
    if (do_prefetch && (sub & 7) == 0) {
        __builtin_prefetch(user + (long)s_src[le + 32] * EMBED_DIM + sub * 4, 0, 0);
    }

    const float4 v = *(const float4*)(user + (long)src * EMBED_DIM + sub * 4);
    float* o = out + (long)dst * EMBED_DIM + sub * 4;
    atomicAdd(o + 0, v.x);
    atomicAdd(o + 1, v.y);
    atomicAdd(o + 2, v.z);
    atomicAdd(o + 3, v.w);
    if (sub == 0) atomicAdd(&counts[dst], 1u);
}

// ---------------------------------------------------------------------------
// Pass 2 (defined FIRST so it leads the disasm): edge scatter. One block = one
// tile of 1024 edges. Edge indices staged into LDS via UNCONDITIONAL async
// b128 loads (source address clamped to NUM_EDGES-4: every in-range LDS slot
// is staged by a thread whose 4-int window fits; out-of-range slots get
// garbage that the bounds-checked loop never reads). Then 8 waves chew
// through 16 edges per iteration.
// ---------------------------------------------------------------------------
__global__ __launch_bounds__(SCATTER_BLOCK)
void ngcf_scatter(const float* __restrict__ user,
                  const int*   __restrict__ esrc,
                  const int*   __restrict__ edst,
                  float*       __restrict__ out,
                  unsigned*    __restrict__ counts) {
    __shared__ int s_src[TILE_EDGES];
    __shared__ int s_dst[TILE_EDGES];

    const int  t     = threadIdx.x;
    const long tbase = (long)blockIdx.x * TILE_EDGES;
    const bool full  = (tbase + TILE_EDGES) <= NUM_EDGES;   // uniform per block

    // --- stage this tile's edge indices into LDS (async DMA path, no branch) ---
    {
        const int  idx = t * 4;                  // 4 ints/thread -> 1024 ints
        long g = tbase + idx;
        if (g > (long)NUM_EDGES - 4) g = (long)NUM_EDGES - 4;   // clamp, no branch
        async_copy_b128(esrc + g, &s_src[idx]);
        async_copy_b128(edst + g, &s_dst[idx]);
    }
    async_wait_all();
    __syncthreads();

    const int sub   = t & 15;   // float4 slot within the 64-float row
    const int eslot = t >> 4;   // which of 16 concurrent edges this lane serves

    if (full) {
        // steady state: no per-iteration bounds checks at all
        #pragma unroll 4
        for (int j = 0; j < TILE_EDGES / 16; ++j) {
            const int le = j * 16 + eslot;
            process_edge(user, out, counts, s_src, s_dst, le, sub,
                         /*do_prefetch=*/le + 32 < TILE_EDGES);
        }
    } else {
        for (int j = 0; j < TILE_EDGES / 16; ++j) {
            const int le = j * 16 + eslot;
            if (tbase + le < NUM_EDGES) {
                process_edge(user, out, counts, s_src, s_dst, le, sub,
                             /*do_prefetch=*/false);
            }
        }
    }
}

// ---------------------------------------------------------------------------
// Pass 1: zero the accumulator (d_out) and the per-item edge counts (d_ws).
// ---------------------------------------------------------------------------
__global__ void ngcf_zero(float4* __restrict__ out4, uint4* __restrict__ cnt4,
                          int n_out4, int n_cnt4) {
    int i = blockIdx.x * blockDim.x + threadIdx.x;
    if (i < n_out4) {
        out4[i] = make_float4(0.f, 0.f, 0.f, 0.f);
    } else {
        int j = i - n_out4;
        if (j < n_cnt4) cnt4[j] = make_uint4(0u, 0u, 0u, 0u);
    }
}

// ---------------------------------------------------------------------------
// Pass 3: out[i,:] /= max(count[i], 1)  (in place, vectorized float4)
// ---------------------------------------------------------------------------
__global__ void ngcf_finalize(float4* __restrict__ out4,
                              const unsigned* __restrict__ counts) {
    const long n = (long)NUM_ITEM * (EMBED_DIM / 4);
    long i = (long)blockIdx.x * blockDim.x + threadIdx.x;
    if (i >= n) return;
    const int      item = (int)(i >> 4);          // 16 float4 per item row
    const unsigned c    = counts[item];
    const float    inv  = 1.0f / (float)(c ? c : 1u);
    float4 v = out4[i];
    v.x *= inv; v.y *= inv; v.z *= inv; v.w *= inv;
    out4[i] = v;
}

// ---------------------------------------------------------------------------
extern "C" void kernel_launch(void* const* d_in, const int* in_sizes, int n_in,
                              void* d_out, int out_size, void* d_ws, size_t ws_size,
                              hipStream_t stream) {
    (void)in_sizes; (void)n_in; (void)out_size; (void)ws_size;

    const float* user  = (const float*)d_in[0];   // [NUM_USER, 64] f32
    /* d_in[1] = item_embed : unused by the reference's net effect */
    const int*   esrc  = (const int*)d_in[2];     // [E] i32
    const int*   edst  = (const int*)d_in[3];     // [E] i32
    float*       out   = (float*)d_out;           // [NUM_ITEM, 64] f32
    unsigned*    counts = (unsigned*)d_ws;        // [NUM_ITEM] u32 (2 MB scratch)

    // Pass 1: zero accumulators + counts
    const int n_out4 = NUM_ITEM * EMBED_DIM / 4;  // 8,000,000
    const int n_cnt4 = NUM_ITEM / 4;              // 125,000
    const int zthreads = n_out4 + n_cnt4;
    ngcf_zero<<<(zthreads + 255) / 256, 256, 0, stream>>>(
        (float4*)out, (uint4*)counts, n_out4, n_cnt4);

    // Pass 2: atomic scatter-sum over edges
    const int ntiles = (NUM_EDGES + TILE_EDGES - 1) / TILE_EDGES;  // 1954
    ngcf_scatter<<<ntiles, SCATTER_BLOCK, 0, stream>>>(user, esrc, edst, out, counts);

    // Pass 3: divide by counts (mean; isolated items stay zero)
    const long nf = (long)NUM_ITEM * (EMBED_DIM / 4);               // 8,000,000
    ngcf_finalize<<<(int)((nf + 255) / 256), 256, 0, stream>>>((float4*)out, counts);
}